// EGNNSiLUModel_55697135894660
// MI455X (gfx1250) — compile-verified
//
#include <hip/hip_runtime.h>
#include <math.h>

#define NNODES 50000
#define NEDGES 800000
#define EMB    64
#define EDGED  32
#define NL     4
#define NG     64

typedef float v2f __attribute__((ext_vector_type(2)));
typedef float v8f __attribute__((ext_vector_type(8)));

__device__ __forceinline__ v8f v8f_zero() {
  v8f z;
#pragma unroll
  for (int i = 0; i < 8; ++i) z[i] = 0.f;
  return z;
}

__device__ __forceinline__ v8f wmma4(v2f a, v2f b, v8f c) {
  // V_WMMA_F32_16X16X4_F32 : D = A(16x4) * B(4x16) + C(16x16), all f32
  return __builtin_amdgcn_wmma_f32_16x16x4_f32(false, a, false, b, (short)0, c,
                                               false, false);
}

__device__ __forceinline__ float silu(float v) {
  return v / (1.f + __expf(-v));
}

// ---------------------------------------------------------------------------
// Column (per-output-feature) sum / sum-of-squares for a 16x64 tile held in
// 4 f32 C accumulators.  Reduces over the 16 rows (8 in-lane + cross-half
// shuffle), then atomically adds into dstSum/dstSq (LDS or global).
// ---------------------------------------------------------------------------
__device__ __forceinline__ void wave_col_stats(const v8f* acc, int lane,
                                               float* dstSum, float* dstSq) {
  const int n = lane & 15, half = lane >> 4;
#pragma unroll
  for (int t = 0; t < 4; ++t) {
    float s = 0.f, q = 0.f;
#pragma unroll
    for (int j = 0; j < 8; ++j) {
      float v = acc[t][j];
      s += v;
      q += v * v;
    }
    s += __shfl_xor(s, 16, 32);
    q += __shfl_xor(q, 16, 32);
    if (half == 0) {
      atomicAdd(&dstSum[16 * t + n], s);
      atomicAdd(&dstSq[16 * t + n], q);
    }
  }
}

// ---------------------------------------------------------------------------
// One branch-free K segment of edge GEMM1.  aBase/Wm/Wc already include the
// per-half +2*half K offset, so every address is base + compile-time k0 and
// folds into the load's immediate offset (no per-iteration control flow).
// ---------------------------------------------------------------------------
template <bool DO_C, int KC>
__device__ __forceinline__ void gemm1_seg(const float* __restrict__ aBase,
                                          const float* __restrict__ Wm,
                                          const float* __restrict__ Wc, int n,
                                          v8f* accm, v8f* accc) {
#pragma unroll
  for (int k0 = 0; k0 < KC; k0 += 4) {
    v2f a = *(const v2f*)(aBase + k0);  // 8B aligned: base even, k0 mult of 4
#pragma unroll
    for (int t = 0; t < 4; ++t) {
      const int col = 16 * t + n;
      const float* wp = Wm + col * 161 + k0;
      v2f b;
      b.x = wp[0];
      b.y = wp[1];
      accm[t] = wmma4(a, b, accm[t]);
      if (DO_C) {
        const float* wq = Wc + col * 161 + k0;
        v2f bc;
        bc.x = wq[0];
        bc.y = wq[1];
        accc[t] = wmma4(a, bc, accc[t]);
      }
    }
  }
}

// ---------------------------------------------------------------------------
// Edge GEMM1: y = m_in @ W1.T + b1 with m_in = [h[dst](64) | h[src](64) |
// r(1) | edge_s(32)].  WMMA covers the 160 "dense" K columns in three fixed
// segments; the single r column is a rank-1 VALU update.
// ---------------------------------------------------------------------------
template <bool DO_C>
__device__ __forceinline__ void edge_gemm1(
    const float* __restrict__ h, const float* __restrict__ posc,
    const float* __restrict__ edge_s, const int* __restrict__ ei, long base,
    int lane, const float* __restrict__ W1m, const float* __restrict__ b1m,
    const float* __restrict__ W1c, const float* __restrict__ b1c, v8f* accm,
    v8f* accc, int& dst_e, int& src_e, float& dx, float& dy, float& dz) {
  const int n = lane & 15, half = lane >> 4;
  const int off2 = 2 * half;
  const long e = base + n;
  src_e = ei[e];
  dst_e = ei[(long)NEDGES + e];
  dx = posc[(long)dst_e * 3 + 0] - posc[(long)src_e * 3 + 0];
  dy = posc[(long)dst_e * 3 + 1] - posc[(long)src_e * 3 + 1];
  dz = posc[(long)dst_e * 3 + 2] - posc[(long)src_e * 3 + 2];
  const float r = sqrtf(dx * dx + dy * dy + dz * dz);

#pragma unroll
  for (int t = 0; t < 4; ++t) {
    accm[t] = v8f_zero();
    if (DO_C) accc[t] = v8f_zero();
  }

  const float* Wc0 = DO_C ? W1c : W1m;  // dead when !DO_C
  // K segment 0: h[dst], weight cols [0,64)
  gemm1_seg<DO_C, 64>(h + (long)dst_e * EMB + off2, W1m + off2, Wc0 + off2, n,
                      accm, accc);
  // K segment 1: h[src], weight cols [64,128)
  gemm1_seg<DO_C, 64>(h + (long)src_e * EMB + off2, W1m + 64 + off2,
                      Wc0 + 64 + off2, n, accm, accc);
  // K segment 2: edge_s, weight cols [129,161) (skip r column at 128)
  gemm1_seg<DO_C, 32>(edge_s + e * EDGED + off2, W1m + 129 + off2,
                      Wc0 + 129 + off2, n, accm, accc);

  // rank-1 r-column + bias
#pragma unroll
  for (int j = 0; j < 8; ++j) {
    const float rj = __shfl(r, j + 8 * half, 32);
#pragma unroll
    for (int t = 0; t < 4; ++t) {
      const int col = 16 * t + n;
      accm[t][j] += rj * W1m[col * 161 + 128];
      if (DO_C) accc[t][j] += rj * W1c[col * 161 + 128];
    }
  }
#pragma unroll
  for (int t = 0; t < 4; ++t) {
    const int col = 16 * t + n;
    const float bm = b1m[col];
    const float bc = DO_C ? b1c[col] : 0.f;
#pragma unroll
    for (int j = 0; j < 8; ++j) {
      accm[t][j] += bm;
      if (DO_C) accc[t][j] += bc;
    }
  }
}

// ---------------------------------------------------------------------------
// Edge GEMM2: y2 = x @ W2.T + b2, K = 64.  x arrives in C-matrix register
// layout; transpose through a per-wave LDS tile (stride 66 to break bank
// conflicts), then feed WMMA A operands back out of LDS.
// ---------------------------------------------------------------------------
__device__ __forceinline__ void edge_gemm2(float* xs, int lane, const v8f* x,
                                           const float* __restrict__ W2,
                                           const float* __restrict__ b2,
                                           v8f* acc2) {
  const int n = lane & 15, half = lane >> 4;
#pragma unroll
  for (int t = 0; t < 4; ++t)
#pragma unroll
    for (int j = 0; j < 8; ++j)
      xs[(j + 8 * half) * 66 + 16 * t + n] = x[t][j];
  __syncthreads();
#pragma unroll
  for (int t = 0; t < 4; ++t) acc2[t] = v8f_zero();
  const float* aBase = xs + n * 66 + 2 * half;
  const float* wBase = W2 + 2 * half;
#pragma unroll
  for (int k0 = 0; k0 < 64; k0 += 4) {
    v2f a;
    a.x = aBase[k0];
    a.y = aBase[k0 + 1];
#pragma unroll
    for (int t = 0; t < 4; ++t) {
      const int col = 16 * t + n;
      v2f b = *(const v2f*)(wBase + col * 64 + k0);
      acc2[t] = wmma4(a, b, acc2[t]);
    }
  }
#pragma unroll
  for (int t = 0; t < 4; ++t) {
    const float bb = b2[16 * t + n];
#pragma unroll
    for (int j = 0; j < 8; ++j) acc2[t][j] += bb;
  }
}

// ===========================================================================
// Kernels
// ===========================================================================

// h = [node_s|node_type] @ lin_in_w.T + b ; init pos/aggr/stats/pool buffers
__global__ void k_init(const float* __restrict__ node_s,
                       const float* __restrict__ node_type,
                       const float* __restrict__ pos,
                       const float* __restrict__ Win,
                       const float* __restrict__ bin, float* h, float* pos_cur,
                       float* aggr_h, float* aggr_p, float* raw, float* hg,
                       float* cnt) {
  const long idx = (long)blockIdx.x * blockDim.x + threadIdx.x;
  const long stride = (long)gridDim.x * blockDim.x;
  for (long i = idx; i < (long)NNODES * 64; i += stride) {
    const long node = i >> 6;
    const int c = (int)(i & 63);
    float acc = bin[c];
#pragma unroll
    for (int k = 0; k < 6; ++k) acc += node_s[node * 6 + k] * Win[c * 8 + k];
#pragma unroll
    for (int k = 0; k < 2; ++k)
      acc += node_type[node * 2 + k] * Win[c * 8 + 6 + k];
    h[i] = acc;
    aggr_h[i] = 0.f;
  }
  for (long i = idx; i < (long)NNODES * 3; i += stride) {
    pos_cur[i] = pos[i];
    aggr_p[i] = 0.f;
  }
  for (long i = idx; i < 512; i += stride) raw[i] = 0.f;
  for (long i = idx; i < (long)NG * 64; i += stride) hg[i] = 0.f;
  for (long i = idx; i < NG; i += stride) cnt[i] = 0.f;
}

// Pass E1: pre-activation stats for msg1 (target 0) and updc1 (target 1)
__global__ __launch_bounds__(256) void k_edge1(
    const float* __restrict__ h, const float* __restrict__ posc,
    const float* __restrict__ edge_s, const int* __restrict__ ei,
    const float* __restrict__ W1m, const float* __restrict__ b1m,
    const float* __restrict__ W1c, const float* __restrict__ b1c, float* raw) {
  __shared__ float sRed[256];
  const int tid = threadIdx.x, lane = tid & 31, wv = tid >> 5;
  sRed[tid] = 0.f;
  __syncthreads();
  const long base = ((long)blockIdx.x * 8 + wv) * 16;
  v8f accm[4], accc[4];
  int de, se;
  float dx, dy, dz;
  edge_gemm1<true>(h, posc, edge_s, ei, base, lane, W1m, b1m, W1c, b1c, accm,
                   accc, de, se, dx, dy, dz);
  wave_col_stats(accm, lane, &sRed[0], &sRed[64]);
  wave_col_stats(accc, lane, &sRed[128], &sRed[192]);
  __syncthreads();
  atomicAdd(&raw[tid], sRed[tid]);
}

// finalize BN batch stats into (scale, shift); re-zero raw accumulators
__global__ void k_finstat(float* raw, float* fin, int tgt,
                          const float* __restrict__ g,
                          const float* __restrict__ be, float count) {
  const int c = threadIdx.x;
  const float s = raw[tgt * 128 + c];
  const float q = raw[tgt * 128 + 64 + c];
  const float mean = s / count;
  const float var = q / count - mean * mean;
  const float istd = rsqrtf(var + 1e-5f);
  const float sc = g[c] * istd;
  fin[tgt * 128 + c] = sc;
  fin[tgt * 128 + 64 + c] = be[c] - mean * sc;
  raw[tgt * 128 + c] = 0.f;
  raw[tgt * 128 + 64 + c] = 0.f;
}

// Pass E2: recompute GEMM1 -> BN+SiLU; coord branch -> aggr_p atomics;
// msg branch GEMM2 pre-act stats (target 2)
__global__ __launch_bounds__(256) void k_edge2(
    const float* __restrict__ h, const float* __restrict__ posc,
    const float* __restrict__ edge_s, const int* __restrict__ ei,
    const float* __restrict__ W1m, const float* __restrict__ b1m,
    const float* __restrict__ W1c, const float* __restrict__ b1c,
    const float* __restrict__ W2m, const float* __restrict__ b2m,
    const float* __restrict__ w2c, const float* __restrict__ b2c,
    const float* __restrict__ fin, float* raw, float* aggr_p) {
  __shared__ float sRed[128];
  __shared__ float xs[8][16 * 66];
  const int tid = threadIdx.x, lane = tid & 31, wv = tid >> 5;
  const int n = lane & 15, half = lane >> 4;
  if (tid < 128) sRed[tid] = 0.f;
  const long base = ((long)blockIdx.x * 8 + wv) * 16;
  v8f accm[4], accc[4];
  int de, se;
  float dx, dy, dz;
  edge_gemm1<true>(h, posc, edge_s, ei, base, lane, W1m, b1m, W1c, b1c, accm,
                   accc, de, se, dx, dy, dz);
  // BN + SiLU in place
#pragma unroll
  for (int t = 0; t < 4; ++t) {
    const int col = 16 * t + n;
    const float sm = fin[col], hm = fin[64 + col];
    const float sc = fin[128 + col], hc = fin[192 + col];
#pragma unroll
    for (int j = 0; j < 8; ++j) {
      accm[t][j] = silu(accm[t][j] * sm + hm);
      accc[t][j] = silu(accc[t][j] * sc + hc);
    }
  }
  // coord branch: s = c . w2c + b ; coord_msg = s * d ; scatter to aggr_p
  float part[8];
#pragma unroll
  for (int j = 0; j < 8; ++j) {
    float p = 0.f;
#pragma unroll
    for (int t = 0; t < 4; ++t) p += accc[t][j] * w2c[16 * t + n];
    part[j] = p;
  }
#pragma unroll
  for (int m = 1; m < 16; m <<= 1)
#pragma unroll
    for (int j = 0; j < 8; ++j) part[j] += __shfl_xor(part[j], m, 32);
  const float bcv = b2c[0];
#pragma unroll
  for (int j = 0; j < 8; ++j) {
    const int row = j + 8 * half;
    const float ddx = __shfl(dx, row, 32);
    const float ddy = __shfl(dy, row, 32);
    const float ddz = __shfl(dz, row, 32);
    const int dr = __shfl(de, row, 32);
    if (n == j) {
      const float s = part[j] + bcv;
      atomicAdd(&aggr_p[(long)dr * 3 + 0], s * ddx);
      atomicAdd(&aggr_p[(long)dr * 3 + 1], s * ddy);
      atomicAdd(&aggr_p[(long)dr * 3 + 2], s * ddz);
    }
  }
  // msg branch GEMM2 (includes __syncthreads covering sRed zero)
  v8f acc2[4];
  edge_gemm2(&xs[wv][0], lane, accm, W2m, b2m, acc2);
  wave_col_stats(acc2, lane, &sRed[0], &sRed[64]);
  __syncthreads();
  if (tid < 128) atomicAdd(&raw[256 + tid], sRed[tid]);
}

// Pass E3: recompute msg path fully, final BN+SiLU, scatter into aggr_h
__global__ __launch_bounds__(256) void k_edge3(
    const float* __restrict__ h, const float* __restrict__ posc,
    const float* __restrict__ edge_s, const int* __restrict__ ei,
    const float* __restrict__ W1m, const float* __restrict__ b1m,
    const float* __restrict__ W2m, const float* __restrict__ b2m,
    const float* __restrict__ fin, float* aggr_h) {
  __shared__ float xs[8][16 * 66];
  const int tid = threadIdx.x, lane = tid & 31, wv = tid >> 5;
  const int n = lane & 15, half = lane >> 4;
  const long base = ((long)blockIdx.x * 8 + wv) * 16;
  v8f accm[4], accc[4];
  int de, se;
  float dx, dy, dz;
  edge_gemm1<false>(h, posc, edge_s, ei, base, lane, W1m, b1m, nullptr,
                    nullptr, accm, accc, de, se, dx, dy, dz);
#pragma unroll
  for (int t = 0; t < 4; ++t) {
    const int col = 16 * t + n;
    const float sm = fin[col], hm = fin[64 + col];
#pragma unroll
    for (int j = 0; j < 8; ++j) accm[t][j] = silu(accm[t][j] * sm + hm);
  }
  v8f acc2[4];
  edge_gemm2(&xs[wv][0], lane, accm, W2m, b2m, acc2);
  int dr[8];
#pragma unroll
  for (int j = 0; j < 8; ++j) dr[j] = __shfl(de, j + 8 * half, 32);
#pragma unroll
  for (int t = 0; t < 4; ++t) {
    const int col = 16 * t + n;
    const float s2 = fin[256 + col], h2 = fin[256 + 64 + col];
#pragma unroll
    for (int j = 0; j < 8; ++j) {
      const float m = silu(acc2[t][j] * s2 + h2);
      atomicAdd(&aggr_h[(long)dr[j] * 64 + col], m);
    }
  }
}

// Node GEMM1: u = [h|aggr_h] @ W1.T + b1 (K=128); store pre-act + stats
__global__ __launch_bounds__(32) void k_node1(
    const float* __restrict__ h, const float* __restrict__ aggr_h,
    const float* __restrict__ W1, const float* __restrict__ b1, float* u_pre,
    float* raw) {
  const int lane = threadIdx.x;
  const int n = lane & 15, half = lane >> 4;
  const int off2 = 2 * half;
  const long base = (long)blockIdx.x * 16;
  v8f acc[4];
#pragma unroll
  for (int t = 0; t < 4; ++t) acc[t] = v8f_zero();
  // K segment 0: h
  {
    const float* aBase = h + (base + n) * 64 + off2;
    const float* wBase = W1 + off2;
#pragma unroll
    for (int k0 = 0; k0 < 64; k0 += 4) {
      v2f a = *(const v2f*)(aBase + k0);
#pragma unroll
      for (int t = 0; t < 4; ++t) {
        const int col = 16 * t + n;
        v2f b = *(const v2f*)(wBase + col * 128 + k0);
        acc[t] = wmma4(a, b, acc[t]);
      }
    }
  }
  // K segment 1: aggr_h
  {
    const float* aBase = aggr_h + (base + n) * 64 + off2;
    const float* wBase = W1 + 64 + off2;
#pragma unroll
    for (int k0 = 0; k0 < 64; k0 += 4) {
      v2f a = *(const v2f*)(aBase + k0);
#pragma unroll
      for (int t = 0; t < 4; ++t) {
        const int col = 16 * t + n;
        v2f b = *(const v2f*)(wBase + col * 128 + k0);
        acc[t] = wmma4(a, b, acc[t]);
      }
    }
  }
#pragma unroll
  for (int t = 0; t < 4; ++t) {
    const int col = 16 * t + n;
    const float bb = b1[col];
#pragma unroll
    for (int j = 0; j < 8; ++j) {
      const int row = j + 8 * half;
      const float v = acc[t][j] + bb;
      acc[t][j] = v;
      u_pre[(base + row) * 64 + col] = v;
    }
  }
  wave_col_stats(acc, lane, &raw[3 * 128], &raw[3 * 128 + 64]);
}

// Node GEMM2: h += BNSiLU(u_pre) @ W2.T + b2 (K=64)
__global__ __launch_bounds__(32) void k_node2(float* h,
                                              const float* __restrict__ u_pre,
                                              const float* __restrict__ fin,
                                              const float* __restrict__ W2,
                                              const float* __restrict__ b2) {
  const int lane = threadIdx.x;
  const int n = lane & 15, half = lane >> 4;
  const int off2 = 2 * half;
  const long base = (long)blockIdx.x * 16;
  v8f acc[4];
#pragma unroll
  for (int t = 0; t < 4; ++t) acc[t] = v8f_zero();
  const float* aBase = u_pre + (base + n) * 64 + off2;
  const float* wBase = W2 + off2;
  const float* scB = fin + 3 * 128 + off2;
  const float* shB = fin + 3 * 128 + 64 + off2;
#pragma unroll
  for (int k0 = 0; k0 < 64; k0 += 4) {
    v2f a;
    a.x = silu(aBase[k0] * scB[k0] + shB[k0]);
    a.y = silu(aBase[k0 + 1] * scB[k0 + 1] + shB[k0 + 1]);
#pragma unroll
    for (int t = 0; t < 4; ++t) {
      const int col = 16 * t + n;
      v2f b = *(const v2f*)(wBase + col * 64 + k0);
      acc[t] = wmma4(a, b, acc[t]);
    }
  }
#pragma unroll
  for (int t = 0; t < 4; ++t) {
    const int col = 16 * t + n;
    const float bb = b2[col];
#pragma unroll
    for (int j = 0; j < 8; ++j) {
      const int row = j + 8 * half;
      const long idx = (base + row) * 64 + col;
      h[idx] += acc[t][j] + bb;
    }
  }
}

// pos += aggr_p ; zero aggregation buffers for next layer
__global__ void k_posz(float* pos_cur, float* aggr_p, float* aggr_h) {
  const long idx = (long)blockIdx.x * blockDim.x + threadIdx.x;
  const long stride = (long)gridDim.x * blockDim.x;
  for (long i = idx; i < (long)NNODES * 3; i += stride) {
    pos_cur[i] += aggr_p[i];
    aggr_p[i] = 0.f;
  }
  for (long i = idx; i < (long)NNODES * 64; i += stride) aggr_h[i] = 0.f;
}

// graph mean-pool accumulation
__global__ void k_pool(const float* __restrict__ h,
                       const int* __restrict__ batch, float* hg, float* cnt) {
  const long idx = (long)blockIdx.x * blockDim.x + threadIdx.x;
  const long stride = (long)gridDim.x * blockDim.x;
  for (long i = idx; i < (long)NNODES * 64; i += stride) {
    const long node = i >> 6;
    const int c = (int)(i & 63);
    const int b = batch[node];
    atomicAdd(&hg[(long)b * 64 + c], h[i]);
    if (c == 0) atomicAdd(&cnt[b], 1.f);
  }
}

__global__ void k_pred(const float* __restrict__ hg,
                       const float* __restrict__ cnt,
                       const float* __restrict__ pw,
                       const float* __restrict__ pb, float* out) {
  const int g = threadIdx.x;
  const float c = fmaxf(cnt[g], 1.f);
  float acc = pb[0];
#pragma unroll
  for (int k = 0; k < 64; ++k) acc += (hg[g * 64 + k] / c) * pw[k];
  out[g] = acc;
}

// ===========================================================================
extern "C" void kernel_launch(void* const* d_in, const int* in_sizes, int n_in,
                              void* d_out, int out_size, void* d_ws,
                              size_t ws_size, hipStream_t stream) {
  (void)in_sizes;
  (void)n_in;
  (void)out_size;
  (void)ws_size;
  const float* node_s = (const float*)d_in[0];
  const float* node_type = (const float*)d_in[1];
  const float* pos = (const float*)d_in[2];
  const int* edge_index = (const int*)d_in[3];
  const float* edge_s = (const float*)d_in[4];
  const int* batch = (const int*)d_in[5];
  const float* lin_in_w = (const float*)d_in[6];
  const float* lin_in_b = (const float*)d_in[7];
  const float* msg_w1 = (const float*)d_in[8];
  const float* msg_b1 = (const float*)d_in[9];
  const float* msg_g1 = (const float*)d_in[10];
  const float* msg_be1 = (const float*)d_in[11];
  const float* msg_w2 = (const float*)d_in[12];
  const float* msg_b2 = (const float*)d_in[13];
  const float* msg_g2 = (const float*)d_in[14];
  const float* msg_be2 = (const float*)d_in[15];
  const float* updf_w1 = (const float*)d_in[16];
  const float* updf_b1 = (const float*)d_in[17];
  const float* updf_g1 = (const float*)d_in[18];
  const float* updf_be1 = (const float*)d_in[19];
  const float* updf_w2 = (const float*)d_in[20];
  const float* updf_b2 = (const float*)d_in[21];
  const float* updc_w1 = (const float*)d_in[22];
  const float* updc_b1 = (const float*)d_in[23];
  const float* updc_g1 = (const float*)d_in[24];
  const float* updc_be1 = (const float*)d_in[25];
  const float* updc_w2 = (const float*)d_in[26];
  const float* updc_b2 = (const float*)d_in[27];
  const float* lin_pred_w = (const float*)d_in[28];
  const float* lin_pred_b = (const float*)d_in[29];
  float* out = (float*)d_out;

  float* ws = (float*)d_ws;
  const size_t oH = 0;
  const size_t oPOS = oH + (size_t)NNODES * 64;
  const size_t oAGH = oPOS + (size_t)NNODES * 3;
  const size_t oAGP = oAGH + (size_t)NNODES * 64;
  const size_t oUP = oAGP + (size_t)NNODES * 3;
  const size_t oRAW = oUP + (size_t)NNODES * 64;
  const size_t oFIN = oRAW + 512;
  const size_t oHG = oFIN + 512;
  const size_t oCNT = oHG + (size_t)NG * 64;
  float* h = ws + oH;
  float* pos_cur = ws + oPOS;
  float* aggr_h = ws + oAGH;
  float* aggr_p = ws + oAGP;
  float* u_pre = ws + oUP;
  float* raw = ws + oRAW;
  float* fin = ws + oFIN;
  float* hg = ws + oHG;
  float* cnt = ws + oCNT;

  const int EB = NEDGES / (8 * 16);  // 6250, exact
  const int NB = NNODES / 16;        // 3125, exact

  k_init<<<4096, 256, 0, stream>>>(node_s, node_type, pos, lin_in_w, lin_in_b,
                                   h, pos_cur, aggr_h, aggr_p, raw, hg, cnt);

  for (int l = 0; l < NL; ++l) {
    const float* W1m = msg_w1 + (size_t)l * 64 * 161;
    const float* B1m = msg_b1 + (size_t)l * 64;
    const float* W1c = updc_w1 + (size_t)l * 64 * 161;
    const float* B1c = updc_b1 + (size_t)l * 64;
    const float* W2m = msg_w2 + (size_t)l * 64 * 64;
    const float* B2m = msg_b2 + (size_t)l * 64;
    const float* W2c = updc_w2 + (size_t)l * 64;  // (1,64)
    const float* B2c = updc_b2 + (size_t)l;       // (1,)
    const float* Wf1 = updf_w1 + (size_t)l * 64 * 128;
    const float* Bf1 = updf_b1 + (size_t)l * 64;
    const float* Wf2 = updf_w2 + (size_t)l * 64 * 64;
    const float* Bf2 = updf_b2 + (size_t)l * 64;

    k_edge1<<<EB, 256, 0, stream>>>(h, pos_cur, edge_s, edge_index, W1m, B1m,
                                    W1c, B1c, raw);
    k_finstat<<<1, 64, 0, stream>>>(raw, fin, 0, msg_g1 + (size_t)l * 64,
                                    msg_be1 + (size_t)l * 64, (float)NEDGES);
    k_finstat<<<1, 64, 0, stream>>>(raw, fin, 1, updc_g1 + (size_t)l * 64,
                                    updc_be1 + (size_t)l * 64, (float)NEDGES);
    k_edge2<<<EB, 256, 0, stream>>>(h, pos_cur, edge_s, edge_index, W1m, B1m,
                                    W1c, B1c, W2m, B2m, W2c, B2c, fin, raw,
                                    aggr_p);
    k_finstat<<<1, 64, 0, stream>>>(raw, fin, 2, msg_g2 + (size_t)l * 64,
                                    msg_be2 + (size_t)l * 64, (float)NEDGES);
    k_edge3<<<EB, 256, 0, stream>>>(h, pos_cur, edge_s, edge_index, W1m, B1m,
                                    W2m, B2m, fin, aggr_h);
    k_node1<<<NB, 32, 0, stream>>>(h, aggr_h, Wf1, Bf1, u_pre, raw);
    k_finstat<<<1, 64, 0, stream>>>(raw, fin, 3, updf_g1 + (size_t)l * 64,
                                    updf_be1 + (size_t)l * 64, (float)NNODES);
    k_node2<<<NB, 32, 0, stream>>>(h, u_pre, fin, Wf2, Bf2);
    k_posz<<<2048, 256, 0, stream>>>(pos_cur, aggr_p, aggr_h);
  }

  k_pool<<<12500, 256, 0, stream>>>(h, batch, hg, cnt);
  k_pred<<<1, 64, 0, stream>>>(hg, cnt, lin_pred_w, lin_pred_b, out);
}